// HippoProj_28243704938787
// MI455X (gfx1250) — compile-verified
//
#include <hip/hip_runtime.h>

// HiPPO-LegT scan on gfx1250 (MI455X), fp32 WMMA 16x16x4, chunked parallel scan.
//
//   c_t = c_{t-1} @ A^T + f_t * B   (2048 channels, L=512, N=64), emit c_t each step.
//
// Augmented-K GEMM per step: c_t = [c_{t-1} | f_t | 0 0 0] @ M_aug,
// M_aug (68x64) = [A^T ; B ; 0 ; 0 ; 0] -> 17 K-slices of 4 for V_WMMA_F32_16X16X4_F32.
// Wave owns 16 channels; M_aug B-operands are loop-invariant (136 VGPRs); the
// D-layout -> A-layout transpose per step bounces through a wave-private 4KB LDS
// tile (DS ops are in-order within a wave => no barriers).
//
// Parallelism: the monolithic scan is only 128 waves (latency bound, vs. the
// 256MB/23.3TB/s ~ 11us memory floor). Linear recurrence => blocked scan:
//   P0: A^64 via 6 LDS squarings (1 block)
//   P1: per-(chtile,chunk) local scan (zero init), save chunk-end state E(k)   [1024 waves]
//   P2: carry(k+1) = E(k) + carry(k) @ (A^T)^64, E(k) loaded into WMMA C-op    [128 waves]
//   P3: re-run each chunk from carry(k), stream all outputs once               [1024 waves]
// 2.05x WMMA work, 8x parallel width, global traffic stays at the 256MB floor.

typedef float v2f __attribute__((ext_vector_type(2)));
typedef float v8f __attribute__((ext_vector_type(8)));

#define CH      2048
#define LSEQ    512
#define NST     64
#define WPB     4          // waves per block
#define LCHUNK  64
#define NCHUNK  8          // LSEQ / LCHUNK
#define NCT     128        // channel tiles (CH / 16)

// workspace layout (in floats)
static constexpr size_t WS_P = 0;                              // A^64, 64x64
static constexpr size_t WS_E = 4096;                           // E(k): NCT*NCHUNK*1024
static constexpr size_t WS_C = 4096 + (size_t)NCT * NCHUNK * 1024;   // carry_in(k)
static constexpr size_t WS_FLOATS = WS_C + (size_t)NCT * NCHUNK * 1024;

// ---------------------------------------------------------------------------
// Phase 0: P = A^64 (row-major), 6 squarings in LDS. Trivial FLOPs, 1 block.
// ---------------------------------------------------------------------------
__global__ __launch_bounds__(256)
void hippo_pow64(const float* __restrict__ Amat, float* __restrict__ ws)
{
    __shared__ float Pa[4096], Pb[4096];
    const int tid = threadIdx.x;
    for (int i = tid; i < 4096; i += 256) Pa[i] = Amat[i];
    __syncthreads();
    float* src = Pa;
    float* dst = Pb;
    for (int it = 0; it < 6; ++it) {
        for (int e = tid * 16; e < tid * 16 + 16; ++e) {
            const int i = e >> 6, j = e & 63;
            float s = 0.0f;
            for (int kk = 0; kk < 64; ++kk)
                s = fmaf(src[i * 64 + kk], src[kk * 64 + j], s);
            dst[e] = s;
        }
        __syncthreads();
        float* tmp = src; src = dst; dst = tmp;
    }
    for (int i = tid; i < 4096; i += 256) ws[WS_P + i] = src[i];
}

// ---------------------------------------------------------------------------
// Phases 1 & 3: chunk-local scan. FIRSTPASS: zero init, save E(k), no output.
//               else: init from carry_in(k), write all outputs.
// ---------------------------------------------------------------------------
template <bool FIRSTPASS>
__global__ __launch_bounds__(32 * WPB)
void hippo_scan_chunk(const float* __restrict__ f_in,   // (2048, 512)
                      const float* __restrict__ Amat,   // (64, 64) row-major
                      const float* __restrict__ Bvec,   // (64,)
                      float* __restrict__ out,          // (512, 2048, 64)
                      float* __restrict__ ws)
{
    __shared__ float smem[WPB * 16 * NST];
    const int tid  = threadIdx.x;
    const int lane = tid & 31;
    const int wv   = tid >> 5;
    const int lrow = lane & 15;
    const int hi   = lane >> 4;
    const int w    = blockIdx.x * WPB + wv;
    const int k    = w & (NCHUNK - 1);      // chunk index
    const int ct   = w >> 3;                // channel tile
    const int ch0  = ct * 16;

    float* sst = &smem[wv * 16 * NST];      // wave-private 16x64 state tile

    if (FIRSTPASS) {
        for (int i = lane; i < 16 * NST; i += 32) sst[i] = 0.0f;
    } else {
        const float* cin = ws + WS_C + ((size_t)(ct * NCHUNK + k)) * 1024;
        for (int i = lane; i < 16 * NST; i += 32) sst[i] = cin[i];
    }

    // Loop-invariant B-operands of M_aug. f32 4x16 slice layout: vgpr0 = rows
    // {4s, 4s+2} (lo/hi lanes), vgpr1 = rows {4s+1, 4s+3}; M_aug[k][n] = A[n][k]
    // makes each pair a contiguous b64 in row-major A.
    v2f bop[17][4];
#pragma unroll
    for (int s = 0; s < 16; ++s) {
#pragma unroll
        for (int j = 0; j < 4; ++j) {
            const int c = 16 * j + lrow;
            bop[s][j] = *(const v2f*)(Amat + c * NST + 4 * s + 2 * hi);
        }
    }
#pragma unroll
    for (int j = 0; j < 4; ++j) {           // slice 16: [B;0;0;0]
        const int c = 16 * j + lrow;
        v2f b16;
        b16.x = hi ? 0.0f : Bvec[c];
        b16.y = 0.0f;
        bop[16][j] = b16;
    }

    const float* frow = f_in + (size_t)(ch0 + lrow) * LSEQ;
    const int t0 = k * LCHUNK, tend = t0 + LCHUNK;
    float4 Fc = *(const float4*)(frow + t0);

    for (int tb = t0; tb < tend; tb += 4) {
        float4 Fn = make_float4(0.0f, 0.0f, 0.0f, 0.0f);
        if (tb + 4 < tend) Fn = *(const float4*)(frow + tb + 4);   // uniform branch

#pragma unroll
        for (int i = 0; i < 4; ++i) {
            const int   t    = tb + i;
            const float fval = (i == 0) ? Fc.x : (i == 1) ? Fc.y
                             : (i == 2) ? Fc.z : Fc.w;

            v8f acc[4] = {};

            // K-slices 0..15: state from LDS in A-operand layout (b64/lane).
#pragma unroll
            for (int s = 0; s < 16; ++s) {
                const v2f aop = *(const v2f*)(sst + lrow * NST + 4 * s + 2 * hi);
#pragma unroll
                for (int j = 0; j < 4; ++j) {
                    acc[j] = __builtin_amdgcn_wmma_f32_16x16x4_f32(
                        false, aop, false, bop[s][j], (short)0, acc[j], false, false);
                }
            }
            // K-slice 16: rank-1 update f_t (x) B folded into the GEMM.
            {
                v2f aop;
                aop.x = hi ? 0.0f : fval;
                aop.y = 0.0f;
#pragma unroll
                for (int j = 0; j < 4; ++j) {
                    acc[j] = __builtin_amdgcn_wmma_f32_16x16x4_f32(
                        false, aop, false, bop[16][j], (short)0, acc[j], false, false);
                }
            }

            // D-layout: vgpr r = rows {r, r+8}, N = lane%16 + 16j.
            float* obase = out + ((size_t)t << 17) + ((size_t)ch0 << 6);
#pragma unroll
            for (int j = 0; j < 4; ++j) {
#pragma unroll
                for (int r = 0; r < 8; ++r) {
                    const int row = r + 8 * hi;
                    const int c   = 16 * j + lrow;
                    const float v = acc[j][r];
                    sst[row * NST + c] = v;
                    if (!FIRSTPASS) obase[row * NST + c] = v;
                }
            }
        }
        Fc = Fn;
    }

    if (FIRSTPASS) {    // save local chunk-end state E(k)
        float* eout = ws + WS_E + ((size_t)(ct * NCHUNK + k)) * 1024;
        for (int i = lane; i < 16 * NST; i += 32) eout[i] = sst[i];
    }
}

// ---------------------------------------------------------------------------
// Phase 2: carry_in(0)=0; carry_in(k+1) = E(k) + carry_in(k) @ (A^T)^64.
// (A^T)^64 column-major == A^64 row-major == ws[WS_P], so the B-operand loads
// are identical in form to the A-matrix loads above. E(k) rides in the WMMA
// C operand (D-layout), so the add costs nothing.
// ---------------------------------------------------------------------------
__global__ __launch_bounds__(32 * WPB)
void hippo_carry(float* __restrict__ ws)
{
    __shared__ float smem[WPB * 16 * NST];
    const int tid  = threadIdx.x;
    const int lane = tid & 31;
    const int wv   = tid >> 5;
    const int lrow = lane & 15;
    const int hi   = lane >> 4;
    const int ct   = blockIdx.x * WPB + wv;

    float* sst = &smem[wv * 16 * NST];
    const float* Pm  = ws + WS_P;
    const float* wsE = ws + WS_E + (size_t)ct * NCHUNK * 1024;
    float*       wsC = ws + WS_C + (size_t)ct * NCHUNK * 1024;

    for (int i = lane; i < 16 * NST; i += 32) { sst[i] = 0.0f; wsC[i] = 0.0f; }

    v2f bop[16][4];
#pragma unroll
    for (int s = 0; s < 16; ++s) {
#pragma unroll
        for (int j = 0; j < 4; ++j) {
            const int c = 16 * j + lrow;
            bop[s][j] = *(const v2f*)(Pm + c * NST + 4 * s + 2 * hi);
        }
    }

    for (int k = 0; k < NCHUNK - 1; ++k) {
        const float* E = wsE + (size_t)k * 1024;
        v8f acc[4];
#pragma unroll
        for (int j = 0; j < 4; ++j) {
#pragma unroll
            for (int r = 0; r < 8; ++r)
                acc[j][r] = E[(r + 8 * hi) * NST + 16 * j + lrow];
        }
#pragma unroll
        for (int s = 0; s < 16; ++s) {
            const v2f aop = *(const v2f*)(sst + lrow * NST + 4 * s + 2 * hi);
#pragma unroll
            for (int j = 0; j < 4; ++j) {
                acc[j] = __builtin_amdgcn_wmma_f32_16x16x4_f32(
                    false, aop, false, bop[s][j], (short)0, acc[j], false, false);
            }
        }
        float* Co = wsC + (size_t)(k + 1) * 1024;
#pragma unroll
        for (int j = 0; j < 4; ++j) {
#pragma unroll
            for (int r = 0; r < 8; ++r) {
                const int row = r + 8 * hi;
                const int c   = 16 * j + lrow;
                const float v = acc[j][r];
                sst[row * NST + c] = v;
                Co[row * NST + c]  = v;
            }
        }
    }
}

// ---------------------------------------------------------------------------
// Fallback: monolithic 512-step scan (128 waves) if workspace is too small.
// ---------------------------------------------------------------------------
__global__ __launch_bounds__(32 * WPB)
void hippo_scan_mono(const float* __restrict__ f_in, const float* __restrict__ Amat,
                     const float* __restrict__ Bvec, float* __restrict__ out)
{
    __shared__ float smem[WPB * 16 * NST];
    const int tid  = threadIdx.x;
    const int lane = tid & 31;
    const int wv   = tid >> 5;
    const int lrow = lane & 15;
    const int hi   = lane >> 4;
    const int ch0  = (blockIdx.x * WPB + wv) * 16;
    float* sst = &smem[wv * 16 * NST];

    for (int i = lane; i < 16 * NST; i += 32) sst[i] = 0.0f;

    v2f bop[17][4];
#pragma unroll
    for (int s = 0; s < 16; ++s)
#pragma unroll
        for (int j = 0; j < 4; ++j)
            bop[s][j] = *(const v2f*)(Amat + (16 * j + lrow) * NST + 4 * s + 2 * hi);
#pragma unroll
    for (int j = 0; j < 4; ++j) {
        v2f b16; b16.x = hi ? 0.0f : Bvec[16 * j + lrow]; b16.y = 0.0f;
        bop[16][j] = b16;
    }

    const float* frow = f_in + (size_t)(ch0 + lrow) * LSEQ;
    float4 Fc = *(const float4*)(frow);
    for (int tb = 0; tb < LSEQ; tb += 4) {
        float4 Fn = make_float4(0.0f, 0.0f, 0.0f, 0.0f);
        if (tb + 4 < LSEQ) Fn = *(const float4*)(frow + tb + 4);
#pragma unroll
        for (int i = 0; i < 4; ++i) {
            const int   t    = tb + i;
            const float fval = (i == 0) ? Fc.x : (i == 1) ? Fc.y
                             : (i == 2) ? Fc.z : Fc.w;
            v8f acc[4] = {};
#pragma unroll
            for (int s = 0; s < 16; ++s) {
                const v2f aop = *(const v2f*)(sst + lrow * NST + 4 * s + 2 * hi);
#pragma unroll
                for (int j = 0; j < 4; ++j)
                    acc[j] = __builtin_amdgcn_wmma_f32_16x16x4_f32(
                        false, aop, false, bop[s][j], (short)0, acc[j], false, false);
            }
            {
                v2f aop; aop.x = hi ? 0.0f : fval; aop.y = 0.0f;
#pragma unroll
                for (int j = 0; j < 4; ++j)
                    acc[j] = __builtin_amdgcn_wmma_f32_16x16x4_f32(
                        false, aop, false, bop[16][j], (short)0, acc[j], false, false);
            }
            float* obase = out + ((size_t)t << 17) + ((size_t)ch0 << 6);
#pragma unroll
            for (int j = 0; j < 4; ++j)
#pragma unroll
                for (int r = 0; r < 8; ++r) {
                    const int row = r + 8 * hi;
                    const int c   = 16 * j + lrow;
                    const float v = acc[j][r];
                    sst[row * NST + c]   = v;
                    obase[row * NST + c] = v;
                }
        }
        Fc = Fn;
    }
}

extern "C" void kernel_launch(void* const* d_in, const int* in_sizes, int n_in,
                              void* d_out, int out_size, void* d_ws, size_t ws_size,
                              hipStream_t stream)
{
    (void)in_sizes; (void)n_in; (void)out_size;
    const float* f_in = (const float*)d_in[0];   // (16,128,512) = (2048,512)
    const float* Amat = (const float*)d_in[1];   // (64,64)
    const float* Bvec = (const float*)d_in[2];   // (64,)
    float* out = (float*)d_out;                  // (512,2048,64)
    float* ws  = (float*)d_ws;

    const dim3 blk(32 * WPB);
    if (ws_size >= WS_FLOATS * sizeof(float)) {
        hipLaunchKernelGGL(hippo_pow64, dim3(1), dim3(256), 0, stream, Amat, ws);
        hipLaunchKernelGGL((hippo_scan_chunk<true>),  dim3(NCT * NCHUNK / WPB), blk, 0, stream,
                           f_in, Amat, Bvec, out, ws);
        hipLaunchKernelGGL(hippo_carry, dim3(NCT / WPB), blk, 0, stream, ws);
        hipLaunchKernelGGL((hippo_scan_chunk<false>), dim3(NCT * NCHUNK / WPB), blk, 0, stream,
                           f_in, Amat, Bvec, out, ws);
    } else {
        hipLaunchKernelGGL(hippo_scan_mono, dim3(NCT / WPB), blk, 0, stream,
                           f_in, Amat, Bvec, out);
    }
}